// DigitCapsuleLayer_11811160064454
// MI455X (gfx1250) — compile-verified
//
#include <hip/hip_runtime.h>

// CapsNet dynamic routing, fused for MI455X (gfx1250, wave32).
//
// Roofline: never materialize u_hat (671MB). b_i is linear in sum of v_j, so each
// of the 3 iterations recomputes u_hat tiles via V_WMMA_F32_16X16X4_F32 from
// inputs (33.5MB) + W (84MB), both L2-resident (192MB) after iter 0.
// Round-2 changes: cross-lane e-dots via DPP8 + permlane16 (VALU) instead of
// ds_bpermute (-320 DS/n); vsum transposed+padded in LDS (80 -> 24 DS/n);
// x tile staged in LDS with coalesced b128 loads (A frags via ds_load_b64).

#define BATCH 64
#define NPRI  16384
#define DPRI  8
#define NCLS  10
#define DCAP  16
#define CPAD  12                           // 10 classes padded to 12 -> 48B rows
#define SVE_ELEMS (BATCH * NCLS * DCAP)    // 10240 floats

#define NWGS   512
#define NCHUNK (NPRI / NWGS)               // 32 primary capsules per WG

typedef __attribute__((ext_vector_type(2))) float v2f;
typedef __attribute__((ext_vector_type(8))) float v8f;

#define DPP8SEL(s0,s1,s2,s3,s4,s5,s6,s7) \
    ((s0)|((s1)<<3)|((s2)<<6)|((s3)<<9)|((s4)<<12)|((s5)<<15)|((s6)<<18)|((s7)<<21))
#define DPP8_XOR1 DPP8SEL(1,0,3,2,5,4,7,6)
#define DPP8_XOR2 DPP8SEL(2,3,0,1,6,7,4,5)
#define DPP8_XOR4 DPP8SEL(4,5,6,7,0,1,2,3)

// Sum across the 16 lanes of a half-wave row, result replicated in all 16.
// xor1/2/4 via DPP8 (8-lane groups), xor8 via v_permlane16_b32 -- all VALU, no DS.
__device__ __forceinline__ float xsum16(float p) {
    p += __int_as_float(__builtin_amdgcn_mov_dpp8(__float_as_int(p), DPP8_XOR1));
    p += __int_as_float(__builtin_amdgcn_mov_dpp8(__float_as_int(p), DPP8_XOR2));
    p += __int_as_float(__builtin_amdgcn_mov_dpp8(__float_as_int(p), DPP8_XOR4));
    int q = __builtin_amdgcn_permlane16(__float_as_int(p), __float_as_int(p),
                                        (int)0xFEDCBA98u, (int)0x76543210u,
                                        false, false);
    p += __int_as_float(q);
    return p;
}

__global__ __launch_bounds__(128)
void caps_iter_kernel(const float* __restrict__ x,    // [B, N, D]
                      const float* __restrict__ W,    // [N, C, D, E]
                      const float* __restrict__ vsum, // [B, C, E]
                      float* __restrict__ s)          // [B, C, E], pre-zeroed
{
    __shared__ float xlds[BATCH * NCHUNK * DPRI];     // 64 KB, x tile for this WG
    __shared__ float vldsT[BATCH * DCAP * CPAD];      // 48 KB, vsum as [b][e][cpad]

    const int tid = threadIdx.x;
    const int n0  = blockIdx.x * NCHUNK;

    // ---- stage vsum transposed: vldsT[(b*16+e)*12 + c] ----
    for (int i = tid; i < SVE_ELEMS; i += 128) {
        const int b = i / (NCLS * DCAP);
        const int c = (i / DCAP) % NCLS;
        const int e = i % DCAP;
        vldsT[(b * DCAP + e) * CPAD + c] = vsum[i];
    }
    // ---- stage x tile, coalesced b128: layout xlds[(b*NCHUNK+nn)*8 + d] ----
    for (int i = tid; i < BATCH * NCHUNK * DPRI / 4; i += 128) {
        const int b  = i >> 6;          // 64 float4 per b (32 n x 2)
        const int no = (i & 63) >> 1;
        const int dp = i & 1;
        ((float4*)xlds)[i] =
            *(const float4*)(x + ((size_t)b * NPRI + n0 + no) * DPRI + dp * 4);
    }
    __syncthreads();

    const int lane = tid & 31;
    const int m    = tid >> 5;          // wave id == batch tile (16 rows)
    const int hi   = lane >> 4;
    const int e    = lane & 15;
    const int dlo  = hi ? 2 : 0;        // A/B K layout: vgpr0={K0|K2}, vgpr1={K1|K3}

    const float* axbase = xlds + (size_t)(m * 16 + e) * NCHUNK * DPRI;
    const int vtbase = ((m * 16 + hi * 8) * DCAP + e) * CPAD;   // + r*16*12

    v8f sacc[NCLS];
    #pragma unroll
    for (int c = 0; c < NCLS; ++c) sacc[c] = v8f{};

    for (int nn = 0; nn < NCHUNK; ++nn) {
        // ---- A fragments from LDS ----
        const float* ax = axbase + nn * DPRI;
        v2f aLo = *(const v2f*)(ax + dlo);
        v2f aHi = *(const v2f*)(ax + dlo + 4);

        const float* wn = W + (size_t)(n0 + nn) * NCLS * DPRI * DCAP;
        __builtin_prefetch(wn + 4 * NCLS * DPRI * DCAP, 0, 1);

        // ---- u_hat tiles, all 10 classes ----
        v8f uh[NCLS];
        #pragma unroll
        for (int c = 0; c < NCLS; ++c) {
            const float* wc = wn + c * DPRI * DCAP;
            v2f bLo, bHi;
            bLo.x = wc[(dlo + 0) * DCAP + e];
            bLo.y = wc[(dlo + 1) * DCAP + e];
            bHi.x = wc[(dlo + 4) * DCAP + e];
            bHi.y = wc[(dlo + 5) * DCAP + e];
            v8f t = v8f{};
            t = __builtin_amdgcn_wmma_f32_16x16x4_f32(false, aLo, false, bLo,
                                                      (short)0, t, false, false);
            t = __builtin_amdgcn_wmma_f32_16x16x4_f32(false, aHi, false, bHi,
                                                      (short)0, t, false, false);
            uh[c] = t;   // element r @ lane: u_hat[16m + r + 8hi, c, e]
        }

        // ---- routing: logits vs vsum, softmax over C, accumulate s ----
        #pragma unroll
        for (int r = 0; r < 8; ++r) {
            const float* vrow = vldsT + vtbase + r * DCAP * CPAD;
            const float4 vA = *(const float4*)(vrow);       // c = 0..3
            const float4 vB = *(const float4*)(vrow + 4);   // c = 4..7
            const float2 vC = *(const float2*)(vrow + 8);   // c = 8..9
            const float vs[NCLS] = { vA.x, vA.y, vA.z, vA.w,
                                     vB.x, vB.y, vB.z, vB.w, vC.x, vC.y };
            float lgt[NCLS];
            #pragma unroll
            for (int c = 0; c < NCLS; ++c)
                lgt[c] = xsum16(uh[c][r] * vs[c]);          // dot over E
            float mx = lgt[0];
            #pragma unroll
            for (int c = 1; c < NCLS; ++c) mx = fmaxf(mx, lgt[c]);
            float den = 0.f;
            #pragma unroll
            for (int c = 0; c < NCLS; ++c) { lgt[c] = __expf(lgt[c] - mx); den += lgt[c]; }
            const float inv = __frcp_rn(den);
            #pragma unroll
            for (int c = 0; c < NCLS; ++c)
                sacc[c][r] += (lgt[c] * inv) * uh[c][r];
        }
    }

    // ---- flush partial s ----
    #pragma unroll
    for (int c = 0; c < NCLS; ++c) {
        #pragma unroll
        for (int r = 0; r < 8; ++r) {
            const int b = m * 16 + hi * 8 + r;
            unsafeAtomicAdd(&s[(b * NCLS + c) * DCAP + e], sacc[c][r]);
        }
    }
}

// squash(s) -> v ; vsum += v (or write final output)
__global__ __launch_bounds__(256)
void caps_squash_kernel(const float* __restrict__ s, float* __restrict__ vsum,
                        float* __restrict__ out, int last)
{
    const int t = blockIdx.x * 256 + threadIdx.x;      // 10240 = 40 * 256
    const float sv = s[t];
    const float sn = xsum16(sv * sv);                  // E=16 aligns with lanes
    const float scale = sn / (1.0f + sn) * __frsqrt_rn(sn + 1e-7f);
    const float v = sv * scale;
    if (last) out[t] = v;
    else      vsum[t] += v;
}

extern "C" void kernel_launch(void* const* d_in, const int* in_sizes, int n_in,
                              void* d_out, int out_size, void* d_ws, size_t ws_size,
                              hipStream_t stream) {
    const float* x = (const float*)d_in[0];   // [64, 16384, 8]
    const float* W = (const float*)d_in[1];   // [16384, 10, 8, 16]
    float* out  = (float*)d_out;              // [64, 10, 16]
    float* s    = (float*)d_ws;               // 40 KB
    float* vsum = s + SVE_ELEMS;              // 40 KB

    hipMemsetAsync(vsum, 0, SVE_ELEMS * sizeof(float), stream);
    for (int it = 0; it < 3; ++it) {
        hipMemsetAsync(s, 0, SVE_ELEMS * sizeof(float), stream);
        caps_iter_kernel<<<NWGS, 128, 0, stream>>>(x, W, vsum, s);
        caps_squash_kernel<<<SVE_ELEMS / 256, 256, 0, stream>>>(s, vsum, out, it == 2);
    }
}